// LoopedTransformer_37349035606833
// MI455X (gfx1250) — compile-verified
//
#include <hip/hip_runtime.h>
#include <hip/hip_bf16.h>
#include <math.h>

// ---------------------------------------------------------------------------
// Looped transformer (B=32,S=128,D=64,H=4,HD=16), CDNA5 / gfx1250.
// One workgroup per batch element; x resident in LDS across all loop steps.
// All GEMMs on V_WMMA_F32_16X16X4_F32 (full f32 precision, matrix pipes).
// ---------------------------------------------------------------------------

typedef __attribute__((ext_vector_type(2))) float v2f;
typedef __attribute__((ext_vector_type(8))) float v8f;

__device__ __forceinline__ v8f wmma4(v2f a, v2f b, v8f c) {
  // D = A(16x4) * B(4x16) + C(16x16), f32, wave32
  return __builtin_amdgcn_wmma_f32_16x16x4_f32(
      /*neg_a=*/false, a, /*neg_b=*/false, b,
      /*c_mod=*/(short)0, c, /*reuse_a=*/false, /*reuse_b=*/false);
}

// ---------------------------------------------------------------------------
// Kernel 1: precompute temb table for every loop step t:
//   te = sinusoidal(t, 256); temb[t] = silu(te @ t_w1 + t_b1) @ t_w2 + t_b2
// ttab: (num_loops, 64) f32 in workspace. One block per t (cap 128 blocks).
// ---------------------------------------------------------------------------
__global__ __launch_bounds__(256) void temb_kernel(
    const int* __restrict__ num_loops,
    const float* __restrict__ t_w1, const float* __restrict__ t_b1,
    const float* __restrict__ t_w2, const float* __restrict__ t_b2,
    float* __restrict__ ttab) {
  const int t = blockIdx.x;
  if (t >= *num_loops) return;
  __shared__ float te[256];
  __shared__ float hs[1024];
  __shared__ float red[256];
  const int tid = threadIdx.x;

  if (tid < 128) {
    float f   = expf(-logf(10000.0f) * (float)tid / 128.0f);
    float arg = (float)t * f;
    te[tid]       = cosf(arg);
    te[128 + tid] = sinf(arg);
  }
  __syncthreads();

  // hs = silu(te @ t_w1 + t_b1); t_w1 row-major (256,1024)
  for (int j = tid; j < 1024; j += 256) {
    float acc = t_b1[j];
    for (int k = 0; k < 256; ++k) acc += te[k] * t_w1[k * 1024 + j];
    hs[j] = acc / (1.0f + expf(-acc));  // x * sigmoid(x)
  }
  __syncthreads();

  // temb = hs @ t_w2 + t_b2; t_w2 row-major (1024,64); 4 partials per output
  const int j = tid & 63, p = tid >> 6;
  float acc = 0.0f;
  for (int k = p * 256; k < p * 256 + 256; ++k) acc += hs[k] * t_w2[k * 64 + j];
  red[tid] = acc;
  __syncthreads();
  if (tid < 64) {
    ttab[t * 64 + tid] =
        t_b2[tid] + red[tid] + red[tid + 64] + red[tid + 128] + red[tid + 192];
  }
}

// ---------------------------------------------------------------------------
// Kernel 2: the full 100-step loop, one block per batch element.
// 512 threads = 16 wave32s. LDS: x | h/att-out | q | k | v  (160 KB).
// ---------------------------------------------------------------------------
#define TB 512
#define NW 16

__device__ __forceinline__ void ln_rows(const float* __restrict__ src,
                                        float* __restrict__ dst,
                                        const float* __restrict__ g,
                                        const float* __restrict__ bta, int tid) {
  // 4 threads per row (rows are 64 wide); reduce within groups of 4 lanes.
  const int s = tid >> 2, p = tid & 3;
  float sum = 0.0f, sq = 0.0f;
  for (int d = p * 16; d < p * 16 + 16; ++d) {
    float xv = src[s * 64 + d];
    sum += xv; sq += xv * xv;
  }
  for (int off = 1; off < 4; off <<= 1) {
    sum += __shfl_xor(sum, off, 4);
    sq  += __shfl_xor(sq,  off, 4);
  }
  const float mu  = sum * (1.0f / 64.0f);
  const float var = sq * (1.0f / 64.0f) - mu * mu;
  const float inv = rsqrtf(var + 1e-5f);
  for (int d = p * 16; d < p * 16 + 16; ++d)
    dst[s * 64 + d] = (src[s * 64 + d] - mu) * inv * g[d] + bta[d];
}

__global__ __launch_bounds__(TB, 1) void xformer_kernel(
    const int* __restrict__ idx, const int* __restrict__ num_loops,
    const float* __restrict__ wte, const float* __restrict__ wpe,
    const float* __restrict__ ln1_g, const float* __restrict__ ln1_b,
    const float* __restrict__ w_qkv, const float* __restrict__ b_qkv,
    const float* __restrict__ w_o, const float* __restrict__ b_o,
    const float* __restrict__ ln2_g, const float* __restrict__ ln2_b,
    const float* __restrict__ w_fc, const float* __restrict__ b_fc,
    const float* __restrict__ w_pr, const float* __restrict__ b_pr,
    const float* __restrict__ lnf_g, const float* __restrict__ lnf_b,
    const float* __restrict__ ttab, float* __restrict__ out) {
  __shared__ float smem[40960];           // 160 KB of the 320 KB/WGP LDS
  float* xb = smem;                        // 128x64 state
  float* hb = smem + 8192;                 // 128x64 LN out / attention out
  float* qb = smem + 16384;                // 128x64
  float* kb = smem + 24576;                // 128x64
  float* vb = smem + 32768;                // 128x64
  float* gb = qb;                          // 128x128 gelu scratch (q+k reuse)
  float* ob = vb;                          // 128x64 MLP partial (v reuse)

  const int b    = blockIdx.x;
  const int tid  = threadIdx.x;
  const int wid  = tid >> 5;
  const int lane = tid & 31;
  const int l16  = lane & 15;
  const int half = lane >> 4;              // 0: lanes 0-15, 1: lanes 16-31
  const int koff = 2 * half;               // wmma f32 A/B lane-half K offset

  // ---- token + position embedding ----
  for (int i = tid; i < 8192; i += TB) {
    int s = i >> 6, d = i & 63;
    xb[i] = wte[idx[b * 128 + s] * 64 + d] + wpe[i];
  }
  __syncthreads();

  const int NL = *num_loops;
  for (int t = 0; t < NL; ++t) {
    // ---- 1. x += temb(t) ----
    for (int i = tid; i < 8192; i += TB) xb[i] += ttab[t * 64 + (i & 63)];
    __syncthreads();

    // ---- 2. LN1 ----
    ln_rows(xb, hb, ln1_g, ln1_b, tid);
    __syncthreads();

    // ---- 3. QKV: h(128x64) @ w_qkv(64x192) + b_qkv ----
    for (int tile = wid; tile < 96; tile += NW) {
      const int m0 = (tile / 12) * 16, n0 = (tile % 12) * 16;
      v8f acc;
      { float bias = b_qkv[n0 + l16]; for (int r = 0; r < 8; ++r) acc[r] = bias; }
      for (int k = 0; k < 64; k += 4) {
        v2f a, w;
        a.x = hb[(m0 + l16) * 64 + k + koff];
        a.y = hb[(m0 + l16) * 64 + k + koff + 1];
        w.x = w_qkv[(k + koff) * 192 + n0 + l16];
        w.y = w_qkv[(k + koff + 1) * 192 + n0 + l16];
        acc = wmma4(a, w, acc);
      }
      float* dst; int nc;
      if (n0 < 64)       { dst = qb; nc = n0; }
      else if (n0 < 128) { dst = kb; nc = n0 - 64; }
      else               { dst = vb; nc = n0 - 128; }
      for (int r = 0; r < 8; ++r)
        dst[(m0 + r + 8 * half) * 64 + nc + l16] = acc[r];
    }
    __syncthreads();

    // ---- 4. attention: job = (head, q-tile), 2 jobs per wave ----
    for (int job = wid; job < 32; job += NW) {
      const int hh = job >> 3, qt = job & 7;
      const int q0 = qt * 16, c0 = hh * 16;
      v8f sc[8];  // full 16x128 score block in registers
      for (int kt = 0; kt < 8; ++kt) {
        v8f a8; for (int r = 0; r < 8; ++r) a8[r] = 0.0f;
        for (int kk = 0; kk < 16; kk += 4) {   // K = HD = 16
          v2f a, bb;
          a.x  = qb[(q0 + l16) * 64 + c0 + kk + koff];
          a.y  = qb[(q0 + l16) * 64 + c0 + kk + koff + 1];
          bb.x = kb[(kt * 16 + l16) * 64 + c0 + kk + koff];  // B = K^T slice
          bb.y = kb[(kt * 16 + l16) * 64 + c0 + kk + koff + 1];
          a8 = wmma4(a, bb, a8);
        }
        for (int r = 0; r < 8; ++r) {          // scale + causal mask
          const int row = q0 + r + 8 * half, col = kt * 16 + l16;
          a8[r] = (col <= row) ? a8[r] * 0.25f : -1e9f;
        }
        sc[kt] = a8;
      }
      // softmax per row: reduce across kt frags + 16-lane half-waves
      for (int r = 0; r < 8; ++r) {
        float mx = -1e30f;
        for (int kt = 0; kt < 8; ++kt) mx = fmaxf(mx, sc[kt][r]);
        for (int off = 8; off; off >>= 1) mx = fmaxf(mx, __shfl_xor(mx, off, 16));
        float sum = 0.0f;
        for (int kt = 0; kt < 8; ++kt) {
          float e = __expf(sc[kt][r] - mx);
          sc[kt][r] = e; sum += e;
        }
        for (int off = 8; off; off >>= 1) sum += __shfl_xor(sum, off, 16);
        const float inv = 1.0f / sum;
        for (int kt = 0; kt < 8; ++kt) sc[kt][r] *= inv;
      }
      // P @ V: relayout P (C/D-frag) -> A-frag via the wave-private out tile
      v8f o; for (int r = 0; r < 8; ++r) o[r] = 0.0f;
      for (int kt = 0; kt < 8; ++kt) {
        for (int r = 0; r < 8; ++r)            // stash P tile (same-wave DS,
          hb[(q0 + r + 8 * half) * 64 + c0 + l16] = sc[kt][r];  // in-order)
        for (int kk = 0; kk < 16; kk += 4) {
          v2f a, bb;
          a.x  = hb[(q0 + l16) * 64 + c0 + kk + koff];
          a.y  = hb[(q0 + l16) * 64 + c0 + kk + koff + 1];
          bb.x = vb[(kt * 16 + kk + koff) * 64 + c0 + l16];
          bb.y = vb[(kt * 16 + kk + koff + 1) * 64 + c0 + l16];
          o = wmma4(a, bb, o);
        }
      }
      for (int r = 0; r < 8; ++r)              // final att-out tile
        hb[(q0 + r + 8 * half) * 64 + c0 + l16] = o[r];
    }
    __syncthreads();

    // ---- 5. x += STEP * (att @ w_o + b_o) ----
    for (int tile = wid; tile < 32; tile += NW) {
      const int m0 = (tile >> 2) * 16, n0 = (tile & 3) * 16;
      v8f acc;
      { float bias = b_o[n0 + l16]; for (int r = 0; r < 8; ++r) acc[r] = bias; }
      for (int k = 0; k < 64; k += 4) {
        v2f a, w;
        a.x = hb[(m0 + l16) * 64 + k + koff];
        a.y = hb[(m0 + l16) * 64 + k + koff + 1];
        w.x = w_o[(k + koff) * 64 + n0 + l16];
        w.y = w_o[(k + koff + 1) * 64 + n0 + l16];
        acc = wmma4(a, w, acc);
      }
      for (int r = 0; r < 8; ++r)
        xb[(m0 + r + 8 * half) * 64 + n0 + l16] += 0.1f * acc[r];
    }
    __syncthreads();

    // ---- 6. LN2 ----
    ln_rows(xb, hb, ln2_g, ln2_b, tid);
    __syncthreads();

    // ---- 7. MLP, hidden dim 256 processed in two halves of 128 ----
    for (int hh2 = 0; hh2 < 2; ++hh2) {
      // fc: gelu(h @ w_fc[:, half] + b_fc) -> gb (128x128)
      for (int tile = wid; tile < 64; tile += NW) {
        const int m0 = (tile >> 3) * 16, ntl = (tile & 7) * 16;
        const int n0g = hh2 * 128 + ntl;
        v8f acc;
        { float bias = b_fc[n0g + l16]; for (int r = 0; r < 8; ++r) acc[r] = bias; }
        for (int k = 0; k < 64; k += 4) {
          v2f a, w;
          a.x = hb[(m0 + l16) * 64 + k + koff];
          a.y = hb[(m0 + l16) * 64 + k + koff + 1];
          w.x = w_fc[(k + koff) * 256 + n0g + l16];
          w.y = w_fc[(k + koff + 1) * 256 + n0g + l16];
          acc = wmma4(a, w, acc);
        }
        for (int r = 0; r < 8; ++r) {          // tanh-approx gelu (jax default)
          float u = acc[r];
          float g = 0.5f * u *
              (1.0f + tanhf(0.7978845608028654f * (u + 0.044715f * u * u * u)));
          gb[(m0 + r + 8 * half) * 128 + ntl + l16] = g;
        }
      }
      __syncthreads();
      // pr: gb(128x128) @ w_pr[half,:](128x64); accumulate across halves
      for (int tile = wid; tile < 32; tile += NW) {
        const int m0 = (tile >> 2) * 16, n0 = (tile & 3) * 16;
        v8f acc;
        if (hh2 == 0) {
          float bias = b_pr[n0 + l16];
          for (int r = 0; r < 8; ++r) acc[r] = bias;
        } else {
          for (int r = 0; r < 8; ++r)
            acc[r] = ob[(m0 + r + 8 * half) * 64 + n0 + l16];
        }
        for (int k = 0; k < 128; k += 4) {
          v2f a, w;
          a.x = gb[(m0 + l16) * 128 + k + koff];
          a.y = gb[(m0 + l16) * 128 + k + koff + 1];
          w.x = w_pr[(hh2 * 128 + k + koff) * 64 + n0 + l16];
          w.y = w_pr[(hh2 * 128 + k + koff + 1) * 64 + n0 + l16];
          acc = wmma4(a, w, acc);
        }
        if (hh2 == 0) {
          for (int r = 0; r < 8; ++r)
            ob[(m0 + r + 8 * half) * 64 + n0 + l16] = acc[r];
        } else {
          for (int r = 0; r < 8; ++r)
            xb[(m0 + r + 8 * half) * 64 + n0 + l16] += 0.1f * acc[r];
        }
      }
      __syncthreads();
    }
  }  // loop over t

  // ---- final LN -> global out ----
  {
    const int s = tid >> 2, p = tid & 3;
    float sum = 0.0f, sq = 0.0f;
    for (int d = p * 16; d < p * 16 + 16; ++d) {
      float xv = xb[s * 64 + d];
      sum += xv; sq += xv * xv;
    }
    for (int off = 1; off < 4; off <<= 1) {
      sum += __shfl_xor(sum, off, 4);
      sq  += __shfl_xor(sq,  off, 4);
    }
    const float mu  = sum * (1.0f / 64.0f);
    const float var = sq * (1.0f / 64.0f) - mu * mu;
    const float inv = rsqrtf(var + 1e-5f);
    for (int d = p * 16; d < p * 16 + 16; ++d)
      out[b * 8192 + s * 64 + d] =
          (xb[s * 64 + d] - mu) * inv * lnf_g[d] + lnf_b[d];
  }
}

// ---------------------------------------------------------------------------
extern "C" void kernel_launch(void* const* d_in, const int* in_sizes, int n_in,
                              void* d_out, int out_size, void* d_ws, size_t ws_size,
                              hipStream_t stream) {
  const int*   idx   = (const int*)  d_in[0];
  const int*   nlp   = (const int*)  d_in[1];
  const float* wte   = (const float*)d_in[2];
  const float* wpe   = (const float*)d_in[3];
  const float* t_w1  = (const float*)d_in[4];
  const float* t_b1  = (const float*)d_in[5];
  const float* t_w2  = (const float*)d_in[6];
  const float* t_b2  = (const float*)d_in[7];
  const float* ln1_g = (const float*)d_in[8];
  const float* ln1_b = (const float*)d_in[9];
  const float* w_qkv = (const float*)d_in[10];
  const float* b_qkv = (const float*)d_in[11];
  const float* w_o   = (const float*)d_in[12];
  const float* b_o   = (const float*)d_in[13];
  const float* ln2_g = (const float*)d_in[14];
  const float* ln2_b = (const float*)d_in[15];
  const float* w_fc  = (const float*)d_in[16];
  const float* b_fc  = (const float*)d_in[17];
  const float* w_pr  = (const float*)d_in[18];
  const float* b_pr  = (const float*)d_in[19];
  const float* lnf_g = (const float*)d_in[20];
  const float* lnf_b = (const float*)d_in[21];

  float* ttab = (float*)d_ws;  // (num_loops, 64) temb table

  temb_kernel<<<128, 256, 0, stream>>>(nlp, t_w1, t_b1, t_w2, t_b2, ttab);
  xformer_kernel<<<32, TB, 0, stream>>>(idx, nlp, wte, wpe, ln1_g, ln1_b,
                                        w_qkv, b_qkv, w_o, b_o, ln2_g, ln2_b,
                                        w_fc, b_fc, w_pr, b_pr, lnf_g, lnf_b,
                                        ttab, (float*)d_out);
}